// MessagePassing_15307263443079
// MI455X (gfx1250) — compile-verified
//
#include <hip/hip_runtime.h>
#include <math.h>

typedef float v2f __attribute__((ext_vector_type(2)));
typedef float v8f __attribute__((ext_vector_type(8)));

__device__ __forceinline__ v8f wmma4(v2f a, v2f b, v8f c) {
  // D = A(16x4 f32) x B(4x16 f32) + C(16x16 f32), wave32
  return __builtin_amdgcn_wmma_f32_16x16x4_f32(false, a, false, b, (short)0, c,
                                               false, false);
}

__device__ __forceinline__ float silu_f(float x) { return x / (1.f + __expf(-x)); }
__device__ __forceinline__ float sigm_f(float x) { return 1.f / (1.f + __expf(-x)); }

// ---------------------------------------------------------------- utility
__global__ void k_fill0(float* p, int n) {
  int i = blockIdx.x * blockDim.x + threadIdx.x;
  if (i < n) p[i] = 0.f;
}

// out[i,o] = attr[i]*scale * sum_c in[i,c] * W[c,o]
__global__ void k_lin_s(float* __restrict__ out, const float* __restrict__ in,
                        const float* __restrict__ W, const float* __restrict__ attr,
                        int n, int din, int dout, float scale) {
  int idx = blockIdx.x * blockDim.x + threadIdx.x;
  if (idx >= n * dout) return;
  int i = idx / dout, o = idx % dout;
  const float* row = in + (size_t)i * din;
  float acc = 0.f;
  for (int c = 0; c < din; ++c) acc += row[c] * W[c * dout + o];
  out[idx] = acc * scale * attr[i];
}

// out[i,o,k] = attr[i]*scale * sum_c in[i,c,k] * W[c,o]   (k = 0..2)
__global__ void k_lin_v(float* __restrict__ out, const float* __restrict__ in,
                        const float* __restrict__ W, const float* __restrict__ attr,
                        int n, int din, int dout, float scale) {
  int idx = blockIdx.x * blockDim.x + threadIdx.x;
  if (idx >= n * dout) return;
  int i = idx / dout, o = idx % dout;
  const float* base = in + (size_t)i * din * 3;
  float a0 = 0.f, a1 = 0.f, a2 = 0.f;
  for (int c = 0; c < din; ++c) {
    float wv = W[c * dout + o];
    a0 += base[c * 3 + 0] * wv;
    a1 += base[c * 3 + 1] * wv;
    a2 += base[c * 3 + 2] * wv;
  }
  float s = scale * attr[i];
  out[(size_t)idx * 3 + 0] = a0 * s;
  out[(size_t)idx * 3 + 1] = a1 * s;
  out[(size_t)idx * 3 + 2] = a2 * s;
}

// out_s = agg_s @ W * scale * attr + sc_s ; then gate (mode 0) or final write (mode 1)
__global__ void k_post_s(const float* __restrict__ agg_s, const float* __restrict__ W,
                         const float* __restrict__ sc_s, const float* __restrict__ attr,
                         float* __restrict__ s_next, float* __restrict__ gtmp,
                         float* __restrict__ dfinal, int n, int mid, int dout,
                         float scale, int mode) {
  int idx = blockIdx.x * blockDim.x + threadIdx.x;
  if (idx >= n * dout) return;
  int i = idx / dout, o = idx % dout;
  const float* row = agg_s + (size_t)i * mid;
  float acc = 0.f;
  for (int c = 0; c < mid; ++c) acc += row[c] * W[c * dout + o];
  float v = acc * scale * attr[i] + sc_s[(size_t)i * dout + o];
  if (mode == 0) {
    if (o < 64) s_next[(size_t)i * 64 + o] = silu_f(v);
    else        gtmp[(size_t)i * 32 + (o - 64)] = sigm_f(v);
  } else {
    dfinal[(size_t)i * 40 + o] = v;
  }
}

__global__ void k_post_v(const float* __restrict__ agg_v, const float* __restrict__ W,
                         const float* __restrict__ sc_v, const float* __restrict__ attr,
                         const float* __restrict__ gtmp, float* __restrict__ v_next,
                         float* __restrict__ dfinal, int n, int mid, int dvout,
                         float scale, int mode) {
  int idx = blockIdx.x * blockDim.x + threadIdx.x;
  if (idx >= n * dvout) return;
  int i = idx / dvout, j = idx % dvout;
  const float* base = agg_v + (size_t)i * mid * 3;
  float a0 = 0.f, a1 = 0.f, a2 = 0.f;
  for (int c = 0; c < mid; ++c) {
    float wv = W[c * dvout + j];
    a0 += base[c * 3 + 0] * wv;
    a1 += base[c * 3 + 1] * wv;
    a2 += base[c * 3 + 2] * wv;
  }
  float s = scale * attr[i];
  float o0 = a0 * s + sc_v[((size_t)i * dvout + j) * 3 + 0];
  float o1 = a1 * s + sc_v[((size_t)i * dvout + j) * 3 + 1];
  float o2 = a2 * s + sc_v[((size_t)i * dvout + j) * 3 + 2];
  if (mode == 0) {
    float g = gtmp[(size_t)i * 32 + j];
    v_next[((size_t)i * dvout + j) * 3 + 0] = o0 * g;
    v_next[((size_t)i * dvout + j) * 3 + 1] = o1 * g;
    v_next[((size_t)i * dvout + j) * 3 + 2] = o2 * g;
  } else {
    dfinal[(size_t)i * 40 + 16 + j * 3 + 0] = o0;
    dfinal[(size_t)i * 40 + 16 + j * 3 + 1] = o1;
    dfinal[(size_t)i * 40 + 16 + j * 3 + 2] = o2;
  }
}

// ------------------------------------------------------------ edge kernel
// One wave = one tile of 16 edges. Edge MLP (8->64->64->NUMEL) done with
// V_WMMA_F32_16X16X4_F32; final GEMM chunked by 16 output cols, each chunk
// scattered straight from the C-fragment with fp32 global atomics.
template <int CS, int CV>
__global__ __launch_bounds__(128) void k_edge(
    const float* __restrict__ escal, const int* __restrict__ esrc,
    const int* __restrict__ edst, const float* __restrict__ eattr,
    const float* __restrict__ s1, const float* __restrict__ v1,
    float* __restrict__ agg_s, float* __restrict__ agg_v,
    const float* __restrict__ W0, const float* __restrict__ W1,
    const float* __restrict__ W2, int E) {
  constexpr int NUMEL = 2 * (CS + CV);
  constexpr int MID = CS + CV;
  constexpr int NCHUNK = NUMEL / 16;

  __shared__ float sW0[8 * 65];          // padded rows: conflict-free
  __shared__ float stage[4][16 * 65];    // per-wave 16x64 staging, stride 65

  for (int i = threadIdx.x; i < 512; i += 128)
    sW0[(i >> 6) * 65 + (i & 63)] = W0[i];
  __syncthreads();

  const int wave = threadIdx.x >> 5;
  const int lane = threadIdx.x & 31;
  const int half = lane >> 4;   // 0: lanes 0-15, 1: lanes 16-31
  const int m = lane & 15;
  const int tile = blockIdx.x * 4 + wave;
  const int e0 = tile * 16;
  if (e0 >= E) return;
  float* st = stage[wave];

  // A-fragments of X (16 edges x 8 edge-scalars), per ISA 16x4 f32 layout:
  // lane<16 holds K = {4t, 4t+1}, lane>=16 holds K = {4t+2, 4t+3} of row m.
  int xrow = e0 + m; if (xrow > E - 1) xrow = E - 1;
  const float* xp = escal + (size_t)xrow * 8 + 2 * half;
  v2f a0; a0.x = xp[0]; a0.y = xp[1];
  v2f a1; a1.x = xp[4]; a1.y = xp[5];

  const float is8 = 0.35355339059327373f;  // 1/sqrt(8)

  // H1 = silu(X @ W0 / sqrt(8)) : 16x64
#pragma unroll
  for (int n = 0; n < 4; ++n) {
    v8f c = {0.f, 0.f, 0.f, 0.f, 0.f, 0.f, 0.f, 0.f};
    v2f b;
    b.x = sW0[(2 * half + 0) * 65 + 16 * n + m];
    b.y = sW0[(2 * half + 1) * 65 + 16 * n + m];
    c = wmma4(a0, b, c);
    b.x = sW0[(4 + 2 * half + 0) * 65 + 16 * n + m];
    b.y = sW0[(4 + 2 * half + 1) * 65 + 16 * n + m];
    c = wmma4(a1, b, c);
#pragma unroll
    for (int r = 0; r < 8; ++r)
      st[(r + 8 * half) * 65 + 16 * n + m] = silu_f(c[r] * is8);
  }

  // H2 = silu(H1 @ W1 / 8) : 16x64 (A-frags re-read from LDS staging)
  v8f h2[4];
#pragma unroll
  for (int n = 0; n < 4; ++n) {
    v8f c = {0.f, 0.f, 0.f, 0.f, 0.f, 0.f, 0.f, 0.f};
#pragma unroll
    for (int kt = 0; kt < 16; ++kt) {
      const int kr = 4 * kt + 2 * half;
      v2f a; a.x = st[m * 65 + kr]; a.y = st[m * 65 + kr + 1];
      v2f b; b.x = W1[kr * 64 + 16 * n + m]; b.y = W1[(kr + 1) * 64 + 16 * n + m];
      c = wmma4(a, b, c);
    }
    h2[n] = c;
  }
#pragma unroll
  for (int n = 0; n < 4; ++n)
#pragma unroll
    for (int r = 0; r < 8; ++r)
      st[(r + 8 * half) * 65 + 16 * n + m] = silu_f(h2[n][r] * 0.125f);

  // w = H2 @ W2 / 8, 16-column chunks; chunk boundaries align with the
  // w1|w2|w3|w4 split, so the message type is uniform per chunk.
  for (int nc = 0; nc < NCHUNK; ++nc) {
    v8f c = {0.f, 0.f, 0.f, 0.f, 0.f, 0.f, 0.f, 0.f};
#pragma unroll
    for (int kt = 0; kt < 16; ++kt) {
      const int kr = 4 * kt + 2 * half;
      v2f a; a.x = st[m * 65 + kr]; a.y = st[m * 65 + kr + 1];
      v2f b; b.x = W2[kr * NUMEL + 16 * nc + m];
      b.y = W2[(kr + 1) * NUMEL + 16 * nc + m];
      c = wmma4(a, b, c);
    }
    const int f = 16 * nc + m;  // output feature column held by this lane
#pragma unroll
    for (int r = 0; r < 8; ++r) {
      const int e = e0 + r + 8 * half;  // edge row held by c[r]
      if (e >= E) continue;
      const float wv = c[r] * 0.125f;
      const int src = esrc[e];
      const int dst = edst[e];
      const float sh0 = eattr[4 * e + 0];
      if (f < CS) {  // m_s_a
        unsafeAtomicAdd(&agg_s[(size_t)dst * MID + f],
                        s1[(size_t)src * CS + f] * sh0 * wv);
      } else if (f < 2 * CS) {  // m_v_a
        const int ch = f - CS;
        const float sv = s1[(size_t)src * CS + ch] * wv;
        float* p = &agg_v[((size_t)dst * MID + ch) * 3];
        unsafeAtomicAdd(p + 0, sv * eattr[4 * e + 1]);
        unsafeAtomicAdd(p + 1, sv * eattr[4 * e + 2]);
        unsafeAtomicAdd(p + 2, sv * eattr[4 * e + 3]);
      } else if (CV > 0 && f < 2 * CS + CV) {  // m_v_b
        const int ch = f - 2 * CS;
        const float* vp = &v1[((size_t)src * CV + ch) * 3];
        float* p = &agg_v[((size_t)dst * MID + CS + ch) * 3];
        const float g = sh0 * wv;
        unsafeAtomicAdd(p + 0, vp[0] * g);
        unsafeAtomicAdd(p + 1, vp[1] * g);
        unsafeAtomicAdd(p + 2, vp[2] * g);
      } else if (CV > 0) {  // m_s_b
        const int ch = f - 2 * CS - CV;
        const float* vp = &v1[((size_t)src * CV + ch) * 3];
        const float dot = vp[0] * eattr[4 * e + 1] + vp[1] * eattr[4 * e + 2] +
                          vp[2] * eattr[4 * e + 3];
        unsafeAtomicAdd(&agg_s[(size_t)dst * MID + CS + ch],
                        dot * 0.5773502691896258f * wv);
      }
    }
  }
}

// ---------------------------------------------------------------- launcher
extern "C" void kernel_launch(void* const* d_in, const int* in_sizes, int n_in,
                              void* d_out, int out_size, void* d_ws, size_t ws_size,
                              hipStream_t stream) {
  (void)out_size; (void)ws_size;
  const float* nf    = (const float*)d_in[0];
  const float* attr  = (const float*)d_in[1];
  const int*   esrc  = (const int*)d_in[2];
  const int*   edst  = (const int*)d_in[3];
  const float* eattr = (const float*)d_in[4];
  const float* escal = (const float*)d_in[5];
  const int N = in_sizes[0] / 16;
  const int E = in_sizes[2];

  // params: pytree dict keys sorted: fc(list), lin1_ws, lin1_wv, lin2_ws,
  // lin2_wv, sc_ws, sc_wv. Layer 0 has two zero-size leaves (lin1_wv, sc_wv)
  // which may or may not be present in d_in.
  const float* P[3][9];
  {
    int idx = 6;
    const bool empties = (n_in >= 33);
    for (int l = 0; l < 3; ++l) {
      P[l][0] = (const float*)d_in[idx++];  // fc0 (8x64)
      P[l][1] = (const float*)d_in[idx++];  // fc1 (64x64)
      P[l][2] = (const float*)d_in[idx++];  // fc2 (64xnumel)
      P[l][3] = (const float*)d_in[idx++];  // lin1_ws
      P[l][4] = (l > 0 || empties) ? (const float*)d_in[idx++] : nullptr;  // lin1_wv
      P[l][5] = (const float*)d_in[idx++];  // lin2_ws
      P[l][6] = (const float*)d_in[idx++];  // lin2_wv
      P[l][7] = (const float*)d_in[idx++];  // sc_ws
      P[l][8] = (l > 0 || empties) ? (const float*)d_in[idx++] : nullptr;  // sc_wv
    }
  }

  float* w = (float*)d_ws;
  size_t off = 0;
  auto wsalloc = [&](size_t ne) { float* p = w + off; off += ne; return p; };
  float* s1    = wsalloc((size_t)N * 64);
  float* v1    = wsalloc((size_t)N * 32 * 3);
  float* sc_s  = wsalloc((size_t)N * 96);
  float* sc_v  = wsalloc((size_t)N * 32 * 3);
  float* agg_s = wsalloc((size_t)N * 96);
  float* agg_v = wsalloc((size_t)N * 96 * 3);
  float* sA    = wsalloc((size_t)N * 64);
  float* vA    = wsalloc((size_t)N * 32 * 3);
  float* sB    = wsalloc((size_t)N * 64);
  float* vB    = wsalloc((size_t)N * 32 * 3);
  float* gtmp  = wsalloc((size_t)N * 32);

  const int TB = 256;
  auto gridFor = [&](long n) { return dim3((unsigned)((n + TB - 1) / TB)); };
  const int ntiles = (E + 15) / 16;
  dim3 egrid((unsigned)((ntiles + 3) / 4));
  float* fout = (float*)d_out;

  const float iS32 = 0.17677669529663687f;  // 1/sqrt(32)

  // ---------------- Layer 0: cs=16, cv=0 -> 96 scalars, 32 vectors ----
  {
    const int cs = 16, mid = 16, cs_out = 96, cv_out = 32;
    k_fill0<<<gridFor((long)N * mid), TB, 0, stream>>>(agg_s, N * mid);
    k_fill0<<<gridFor((long)N * mid * 3), TB, 0, stream>>>(agg_v, N * mid * 3);
    k_fill0<<<gridFor((long)N * cv_out * 3), TB, 0, stream>>>(sc_v, N * cv_out * 3);
    k_lin_s<<<gridFor((long)N * cs), TB, 0, stream>>>(s1, nf, P[0][3], attr, N, cs, cs, 0.25f);
    k_lin_s<<<gridFor((long)N * cs_out), TB, 0, stream>>>(sc_s, nf, P[0][7], attr, N, cs, cs_out, 0.25f);
    k_edge<16, 0><<<egrid, 128, 0, stream>>>(escal, esrc, edst, eattr, s1, nullptr,
                                             agg_s, agg_v, P[0][0], P[0][1], P[0][2], E);
    const float psc = 0.0625f;  // 1/(sqrt(16)*sqrt(16))
    k_post_s<<<gridFor((long)N * cs_out), TB, 0, stream>>>(agg_s, P[0][5], sc_s, attr,
                                                           sA, gtmp, nullptr, N, mid, cs_out, psc, 0);
    k_post_v<<<gridFor((long)N * cv_out), TB, 0, stream>>>(agg_v, P[0][6], sc_v, attr,
                                                           gtmp, vA, nullptr, N, mid, cv_out, psc, 0);
  }
  // ---------------- Layer 1: cs=64, cv=32 -> 96 scalars, 32 vectors ----
  {
    const int cs = 64, cv = 32, mid = 96, cs_out = 96, cv_out = 32;
    k_fill0<<<gridFor((long)N * mid), TB, 0, stream>>>(agg_s, N * mid);
    k_fill0<<<gridFor((long)N * mid * 3), TB, 0, stream>>>(agg_v, N * mid * 3);
    k_lin_s<<<gridFor((long)N * cs), TB, 0, stream>>>(s1, sA, P[1][3], attr, N, cs, cs, 0.125f);
    k_lin_v<<<gridFor((long)N * cv), TB, 0, stream>>>(v1, vA, P[1][4], attr, N, cv, cv, iS32);
    k_lin_s<<<gridFor((long)N * cs_out), TB, 0, stream>>>(sc_s, sA, P[1][7], attr, N, cs, cs_out, 0.125f);
    k_lin_v<<<gridFor((long)N * cv_out), TB, 0, stream>>>(sc_v, vA, P[1][8], attr, N, cv, cv_out, iS32);
    k_edge<64, 32><<<egrid, 128, 0, stream>>>(escal, esrc, edst, eattr, s1, v1,
                                              agg_s, agg_v, P[1][0], P[1][1], P[1][2], E);
    const float psc = 0.025515518153991437f;  // 1/(sqrt(96)*4)
    k_post_s<<<gridFor((long)N * cs_out), TB, 0, stream>>>(agg_s, P[1][5], sc_s, attr,
                                                           sB, gtmp, nullptr, N, mid, cs_out, psc, 0);
    k_post_v<<<gridFor((long)N * cv_out), TB, 0, stream>>>(agg_v, P[1][6], sc_v, attr,
                                                           gtmp, vB, nullptr, N, mid, cv_out, psc, 0);
  }
  // ---------------- Layer 2: cs=64, cv=32 -> 16 scalars, 8 vectors -----
  {
    const int cs = 64, cv = 32, mid = 96, cs_out = 16, cv_out = 8;
    k_fill0<<<gridFor((long)N * mid), TB, 0, stream>>>(agg_s, N * mid);
    k_fill0<<<gridFor((long)N * mid * 3), TB, 0, stream>>>(agg_v, N * mid * 3);
    k_lin_s<<<gridFor((long)N * cs), TB, 0, stream>>>(s1, sB, P[2][3], attr, N, cs, cs, 0.125f);
    k_lin_v<<<gridFor((long)N * cv), TB, 0, stream>>>(v1, vB, P[2][4], attr, N, cv, cv, iS32);
    k_lin_s<<<gridFor((long)N * cs_out), TB, 0, stream>>>(sc_s, sB, P[2][7], attr, N, cs, cs_out, 0.125f);
    k_lin_v<<<gridFor((long)N * cv_out), TB, 0, stream>>>(sc_v, vB, P[2][8], attr, N, cv, cv_out, iS32);
    k_edge<64, 32><<<egrid, 128, 0, stream>>>(escal, esrc, edst, eattr, s1, v1,
                                              agg_s, agg_v, P[2][0], P[2][1], P[2][2], E);
    const float psc = 0.025515518153991437f;  // 1/(sqrt(96)*4)
    k_post_s<<<gridFor((long)N * cs_out), TB, 0, stream>>>(agg_s, P[2][5], sc_s, attr,
                                                           nullptr, nullptr, fout, N, mid, cs_out, psc, 1);
    k_post_v<<<gridFor((long)N * cv_out), TB, 0, stream>>>(agg_v, P[2][6], sc_v, attr,
                                                           nullptr, nullptr, fout, N, mid, cv_out, psc, 1);
  }
}